// AttentionTopK_27118423507289
// MI455X (gfx1250) — compile-verified
//
#include <hip/hip_runtime.h>
#include <hip/hip_bf16.h>

// ---------------------------------------------------------------------------
// AttentionTopK on MI455X (gfx1250, wave32)
//
// Roofline: must stream x (1 GiB f32) once for scores + 128 MiB for gather.
// ~51 us floor at 23.3 TB/s; compute is ~0.5 GFLOP -> purely bandwidth bound.
// Matvec kept in f32 and mapped onto V_WMMA_F32_16X16X4_F32 (B = W broadcast
// across columns so every D column carries the score). Global loads stay
// coalesced float4; WMMA operands come from an LDS tile (stride-36 padding:
// row*36 mod 64 distinct for r=0..15 -> conflict-free A reads).
// ---------------------------------------------------------------------------

#define BATCH 64
#define T_DIM 8192
#define F_DIM 512
#define TOPK  512

typedef __attribute__((ext_vector_type(2))) float v2f;
typedef __attribute__((ext_vector_type(8))) float v8f;

// ---------------------------------------------------------------------------
// Pass 1: scores[b][t] = tanh(dot(x[b][t][:], W) + b[t])
// 8 waves / block, 16 timesteps per wave -> 128 t's per block, 4096 blocks.
// ---------------------------------------------------------------------------
__global__ __launch_bounds__(256) void attn_score_kernel(
    const float* __restrict__ x, const float* __restrict__ W,
    const float* __restrict__ bvec, float* __restrict__ scores) {
  __shared__ float Wl[F_DIM];
  __shared__ __align__(16) float tile[8][16][36];  // padded: conflict-free rows

  const int tid  = threadIdx.x;
  const int w    = tid >> 5;   // wave in block
  const int lane = tid & 31;

  // Stage W once per block (tiny, 2 KiB)
  Wl[tid]       = W[tid];
  Wl[tid + 256] = W[tid + 256];

  const int batch = blockIdx.x >> 6;
  const int t0    = ((blockIdx.x & 63) << 7) + (w << 4);  // 16 rows per wave
  const float* xrow = x + ((size_t)batch * T_DIM + t0) * F_DIM;

  const int rl = lane >> 3;         // 0..3   (fill: row group)
  const int cl = (lane & 7) << 2;   // 0..28  (fill: col, float4 granules)
  const int h  = lane >> 4;         // 0/1    (WMMA half-wave)
  const int r  = lane & 15;         // WMMA A row (M)

  v8f acc = {0.f, 0.f, 0.f, 0.f, 0.f, 0.f, 0.f, 0.f};
  __syncthreads();

  for (int kb = 0; kb < F_DIM; kb += 32) {
    // Coalesced fill of a 16x32 tile: 32 lanes x float4 x 4 iters
#pragma unroll
    for (int i = 0; i < 4; ++i) {
      const int row = rl + (i << 2);
      const float4 v =
          *(const float4*)(xrow + (size_t)row * F_DIM + kb + cl);
      *(float4*)&tile[w][row][cl] = v;
    }
    if (kb + 32 < F_DIM)  // global_prefetch_b8 of the next k-tile
      __builtin_prefetch(xrow + (size_t)rl * F_DIM + kb + 32 + cl, 0, 3);
    __syncthreads();

    // 8 WMMA steps cover k = kb..kb+31.  A lane layout: K = vgpr + 2*(lane/16).
#pragma unroll
    for (int s = 0; s < 8; ++s) {
      const int c  = (s << 2) + (h << 1);      // tile column of K-pair
      const int wk = kb + c;                   // matching W element
      v2f a, bm;
      a.x  = tile[w][r][c];
      a.y  = tile[w][r][c + 1];
      bm.x = Wl[wk];        // B[k][n] = W[k] for all n -> all D cols = score
      bm.y = Wl[wk + 1];
      acc = __builtin_amdgcn_wmma_f32_16x16x4_f32(
          false, a, false, bm, (short)0, acc, false, false);
    }
    __syncthreads();
  }

  // D layout: vgpr j, lane L -> row M = j + 8*(L/16), col N = L%16.
  // Every column holds the same score; lanes with N==0 write 16 rows/wave.
  if ((lane & 15) == 0) {
#pragma unroll
    for (int j = 0; j < 8; ++j) {
      const int t = t0 + j + (h << 3);
      scores[(size_t)batch * T_DIM + t] = tanhf(acc[j] + bvec[t]);
    }
  }
}

// ---------------------------------------------------------------------------
// Pass 2: per-batch full bitonic sort of 8192 (desc score, ties by asc index),
// then softmax over the leading 512. One 1024-thread block per batch.
// key = ~monotonic(f32): ascending key order == descending score order.
// ---------------------------------------------------------------------------
__global__ __launch_bounds__(1024) void attn_topk_kernel(
    const float* __restrict__ scores, float* __restrict__ weights_out,
    int* __restrict__ idx_out) {
  __shared__ unsigned int   key[T_DIM];    // 32 KiB
  __shared__ unsigned short sidx[T_DIM];   // 16 KiB
  __shared__ float          red[TOPK];     // softmax reduction
  __shared__ float          smax;

  const int tid   = threadIdx.x;
  const int batch = blockIdx.x;
  const float* sc = scores + (size_t)batch * T_DIM;

  for (int i = tid; i < T_DIM; i += 1024) {
    unsigned int u = __float_as_uint(sc[i]);
    u ^= (u >> 31) ? 0xFFFFFFFFu : 0x80000000u;  // monotonic ascending map
    key[i]  = ~u;                                // invert -> descending score
    sidx[i] = (unsigned short)i;
  }
  __syncthreads();

  for (int k = 2; k <= T_DIM; k <<= 1) {
    for (int j = k >> 1; j > 0; j >>= 1) {
      for (int p = tid; p < (T_DIM >> 1); p += 1024) {
        const int l = ((p & ~(j - 1)) << 1) | (p & (j - 1));
        const int m = l | j;
        const bool up = ((l & k) == 0);
        const unsigned int   ka = key[l], kb2 = key[m];
        const unsigned short ia = sidx[l], ib = sidx[m];
        const bool agtb = (ka > kb2) || (ka == kb2 && ia > ib);
        if (agtb == up) {
          key[l] = kb2; key[m] = ka;
          sidx[l] = ib; sidx[m] = ia;
        }
      }
      __syncthreads();
    }
  }

  // Leading TOPK entries are the answer in order. Softmax their raw scores.
  float v = 0.f;
  if (tid < TOPK) {
    const int gi = sidx[tid];
    v = sc[gi];
    idx_out[batch * TOPK + tid] = gi;
  }
  if (tid == 0) smax = sc[sidx[0]];  // rank-0 score == max
  __syncthreads();

  float e = 0.f;
  if (tid < TOPK) {
    e = expf(v - smax);
    red[tid] = e;
  }
  __syncthreads();
  for (int s = TOPK >> 1; s > 0; s >>= 1) {
    if (tid < s) red[tid] += red[tid + s];
    __syncthreads();
  }
  if (tid < TOPK) weights_out[batch * TOPK + tid] = e / red[0];
}

// ---------------------------------------------------------------------------
// Pass 3: gather features: out[b][k][:] = x[b][idx[b][k]][:].
// One wave per (b,k) row; 4x float4 per lane, fully coalesced. 128 MiB total.
// ---------------------------------------------------------------------------
__global__ __launch_bounds__(256) void attn_gather_kernel(
    const float* __restrict__ x, const int* __restrict__ idx,
    float* __restrict__ feat_out) {
  const int gw    = (int)((blockIdx.x * 256u + threadIdx.x) >> 5);  // row id
  const int lane  = threadIdx.x & 31;
  const int batch = gw >> 9;       // / TOPK
  const int kk    = gw & (TOPK - 1);
  const int t     = idx[batch * TOPK + kk];

  const float4* src =
      (const float4*)(x + ((size_t)batch * T_DIM + t) * F_DIM);
  float4* dst =
      (float4*)(feat_out + ((size_t)batch * TOPK + kk) * F_DIM);
#pragma unroll
  for (int i = 0; i < 4; ++i) dst[lane + 32 * i] = src[lane + 32 * i];
}

// ---------------------------------------------------------------------------
extern "C" void kernel_launch(void* const* d_in, const int* in_sizes, int n_in,
                              void* d_out, int out_size, void* d_ws,
                              size_t ws_size, hipStream_t stream) {
  const float* x  = (const float*)d_in[0];  // (64, 8192, 512) f32
  const float* W  = (const float*)d_in[1];  // (512, 1) f32
  const float* bv = (const float*)d_in[2];  // (8192, 1) f32

  float* out     = (float*)d_out;
  float* feat    = out;                                  // (B, K, F)
  float* weights = out + (size_t)BATCH * TOPK * F_DIM;   // (B, K, 1)

  float* scores = (float*)d_ws;                                  // 2 MiB
  int*   idxs   = (int*)((char*)d_ws +
                         (size_t)BATCH * T_DIM * sizeof(float)); // 128 KiB

  attn_score_kernel<<<(BATCH * T_DIM) / 128, 256, 0, stream>>>(x, W, bv,
                                                               scores);
  attn_topk_kernel<<<BATCH, 1024, 0, stream>>>(scores, weights, idxs);
  attn_gather_kernel<<<(BATCH * TOPK) / 8, 256, 0, stream>>>(x, idxs, feat);
}